// DCRNN_RecurrentGCN_16192026706532
// MI455X (gfx1250) — compile-verified
//
#include <hip/hip_runtime.h>
#include <hip/hip_bf16.h>

#define N_NODES 50000
#define N_EDGES 800000
#define LAGSC   16
#define FILTC   64
#define CINC    80
#define TILES   (N_NODES / 16)   // 3125 exactly

typedef float v2f __attribute__((ext_vector_type(2)));
typedef float v8f __attribute__((ext_vector_type(8)));

// ---------------- zero workspace ----------------
__global__ void k_zero(float* __restrict__ p, int n) {
    int i = blockIdx.x * blockDim.x + threadIdx.x;
    if (i < n) p[i] = 0.0f;
}

// ---------------- weighted degrees ----------------
__global__ void k_deg(const int* __restrict__ ei, const float* __restrict__ ew,
                      float* __restrict__ deg_out, float* __restrict__ deg_in) {
    int e = blockIdx.x * blockDim.x + threadIdx.x;
    if (e >= N_EDGES) return;
    int r = ei[e];
    int c = ei[N_EDGES + e];
    float w = ew[e];
    atomicAdd(&deg_out[r], w);
    atomicAdd(&deg_in[c], w);
}

__global__ void k_inv(float* __restrict__ deg_out, float* __restrict__ deg_in) {
    int i = blockIdx.x * blockDim.x + threadIdx.x;
    if (i >= N_NODES) return;
    float d = deg_out[i]; deg_out[i] = (d > 0.0f) ? 1.0f / d : 1.0f;
    d = deg_in[i];        deg_in[i]  = (d > 0.0f) ? 1.0f / d : 1.0f;
}

// ---------------- combine weights: H0==0 -> only first 16 input rows matter.
// Stored TRANSPOSED as Wt[f][k] (64 x 48) so each WMMA B fragment
// (two K-adjacent scalars for one output column f) is one aligned b64 load.
// Combined K rows:
//   k 0..15 : W[0,0][:16] + W[1,0][:16]   (identity hops, both dirs)
//   k 16..31: W[0,1][:16]                 (out-diffusion hop)
//   k 32..47: W[1,1][:16]                 (in-diffusion hop)
// src layout (2,2,80,64): flat ((d*2+k)*80 + c)*64 + f
__global__ void k_wprep(const float* __restrict__ wz, const float* __restrict__ wh,
                        float* __restrict__ Wzt, float* __restrict__ Wht) {
    int i = blockIdx.x * blockDim.x + threadIdx.x;
    if (i >= 2 * 48 * 64) return;
    const float* src = (i < 48 * 64) ? wz : wh;
    float* dst       = (i < 48 * 64) ? Wzt : Wht;
    int j  = i % (48 * 64);
    int rc = j >> 6;   // combined input row (K) 0..47
    int f  = j & 63;   // output filter
    float v;
    if (rc < 16) {
        int c = rc;                                     // d0k0 + d1k0
        v = src[(0 * CINC + c) * FILTC + f] + src[(2 * CINC + c) * FILTC + f];
    } else if (rc < 32) {
        int c = rc - 16;                                // d0k1
        v = src[(1 * CINC + c) * FILTC + f];
    } else {
        int c = rc - 32;                                // d1k1
        v = src[(3 * CINC + c) * FILTC + f];
    }
    dst[f * 48 + rc] = v;                               // transposed store
}

// ---------------- edge gather-scale-scatter (the real bottleneck) ------------
// one thread per (edge, channel)
__global__ void k_scatter(const int* __restrict__ ei, const float* __restrict__ x,
                          const float* __restrict__ inv_out, const float* __restrict__ inv_in,
                          float* __restrict__ Txo, float* __restrict__ Txi) {
    int gid = blockIdx.x * blockDim.x + threadIdx.x;
    if (gid >= N_EDGES * 16) return;
    int e = gid >> 4;
    int c = gid & 15;
    int r  = ei[e];
    int cl = ei[N_EDGES + e];
    // out-direction: row -> col, scaled by 1/deg_out[row], accumulated at col
    atomicAdd(&Txo[cl * 16 + c], x[r * 16 + c] * inv_out[r]);
    // in-direction: col -> row, scaled by 1/deg_in[col], accumulated at row
    atomicAdd(&Txi[r * 16 + c], x[cl * 16 + c] * inv_in[cl]);
}

// ---------------- fused WMMA GRU-cell + readout ------------------------------
// One wave32 per 16-node tile.
// D = [x | Txo | Txi] (16x48) @ Wc (48x64) via 12 K-slices of f32 16x16x4 WMMA,
// for both z and h_tilde; epilogue fused with the lin_w reduction.
__global__ __launch_bounds__(256)
void k_cell(const float* __restrict__ x, const float* __restrict__ Txo,
            const float* __restrict__ Txi,
            const float* __restrict__ Wzt, const float* __restrict__ Wht,
            const float* __restrict__ bz, const float* __restrict__ bh,
            const float* __restrict__ lw, const float* __restrict__ lb,
            float* __restrict__ out) {
    int wave = threadIdx.x >> 5;
    int lane = threadIdx.x & 31;
    int tile = blockIdx.x * 8 + wave;
    if (tile >= TILES) return;          // wave-uniform: EXEC stays all-ones
    int m  = lane & 15;                  // row within A tile / column within D
    int hi = lane >> 4;                  // 0: K0/K1  1: K2/K3 (per ISA A layout)
    int nodeBase = tile * 16;

    const float* rowX = x   + (nodeBase + m) * 16;
    const float* rowO = Txo + (nodeBase + m) * 16;
    const float* rowI = Txi + (nodeBase + m) * 16;

    // A fragments: aF[part][ks] holds A[M=m][4*ks + 2*hi + {0,1}]
    v2f aF[3][4];
#pragma unroll
    for (int ks = 0; ks < 4; ++ks) {
        int k0 = ks * 4 + hi * 2;
        aF[0][ks] = *(const v2f*)(rowX + k0);
        aF[1][ks] = *(const v2f*)(rowO + k0);
        aF[2][ks] = *(const v2f*)(rowI + k0);
    }

    v8f zacc[4], hacc[4];
#pragma unroll
    for (int nt = 0; nt < 4; ++nt) {
        zacc[nt] = (v8f){0.f, 0.f, 0.f, 0.f, 0.f, 0.f, 0.f, 0.f};
        hacc[nt] = (v8f){0.f, 0.f, 0.f, 0.f, 0.f, 0.f, 0.f, 0.f};
    }

#pragma unroll
    for (int nt = 0; nt < 4; ++nt) {
        int n = nt * 16 + m;             // output filter column for this lane
        const float* wzRow = Wzt + n * 48;   // Wt[f][k], contiguous in k
        const float* whRow = Wht + n * 48;
#pragma unroll
        for (int part = 0; part < 3; ++part) {
#pragma unroll
            for (int ks = 0; ks < 4; ++ks) {
                int kb = part * 16 + ks * 4 + hi * 2;  // B rows mirror A K layout
                v2f bzf = *(const v2f*)(wzRow + kb);   // one b64 load
                v2f bhf = *(const v2f*)(whRow + kb);   // one b64 load
                zacc[nt] = __builtin_amdgcn_wmma_f32_16x16x4_f32(
                    false, aF[part][ks], false, bzf, (short)0, zacc[nt], false, false);
                hacc[nt] = __builtin_amdgcn_wmma_f32_16x16x4_f32(
                    false, aF[part][ks], false, bhf, (short)0, hacc[nt], false, false);
            }
        }
    }

    // Epilogue: h = relu((1 - sigmoid(z+bz)) * tanh(ht+bh)); out = h @ lin_w + lin_b
    // C/D layout: VGPR r -> node M=r (lanes 0-15) / M=r+8 (lanes 16-31), col N=m.
    float lbv = lb[0];
    float s[8];
#pragma unroll
    for (int r = 0; r < 8; ++r) s[r] = 0.0f;
#pragma unroll
    for (int nt = 0; nt < 4; ++nt) {
        int f = nt * 16 + m;
        float bzv = bz[f], bhv = bh[f], lwv = lw[f];
#pragma unroll
        for (int r = 0; r < 8; ++r) {
            float z  = 1.0f / (1.0f + __expf(-(zacc[nt][r] + bzv)));
            float ht = tanhf(hacc[nt][r] + bhv);
            float v  = (1.0f - z) * ht;
            v = v > 0.0f ? v : 0.0f;
            s[r] += v * lwv;
        }
    }
#pragma unroll
    for (int r = 0; r < 8; ++r) {
        float sv = s[r];
#pragma unroll
        for (int off = 8; off > 0; off >>= 1)
            sv += __shfl_xor(sv, off, 16);   // reduce 64 filters within each half-wave
        if (m == 0)
            out[nodeBase + r + hi * 8] = sv + lbv;
    }
}

extern "C" void kernel_launch(void* const* d_in, const int* in_sizes, int n_in,
                              void* d_out, int out_size, void* d_ws, size_t ws_size,
                              hipStream_t stream) {
    const float* x  = (const float*)d_in[0];
    const int*   ei = (const int*)d_in[1];     // edge_index (2, E)
    const float* ew = (const float*)d_in[2];
    const float* wz = (const float*)d_in[3];
    const float* bz = (const float*)d_in[4];
    // d_in[5] = w_r, d_in[6] = b_r: dead (R only multiplies H0 == 0)
    const float* wh = (const float*)d_in[7];
    const float* bh = (const float*)d_in[8];
    const float* lw = (const float*)d_in[9];
    const float* lb = (const float*)d_in[10];

    float* ws      = (float*)d_ws;
    float* deg_out = ws;                       // N (becomes inv_deg_out)
    float* deg_in  = ws + N_NODES;             // N (becomes inv_deg_in)
    float* Txo     = ws + 2 * N_NODES;         // N*16
    float* Txi     = Txo + N_NODES * 16;       // N*16
    float* Wzt     = Txi + N_NODES * 16;       // 64*48 (transposed combined)
    float* Wht     = Wzt + 48 * 64;            // 64*48

    int zn = 2 * N_NODES + 32 * N_NODES;       // degrees + Tx buffers
    k_zero<<<(zn + 255) / 256, 256, 0, stream>>>(ws, zn);
    k_deg<<<(N_EDGES + 255) / 256, 256, 0, stream>>>(ei, ew, deg_out, deg_in);
    k_inv<<<(N_NODES + 255) / 256, 256, 0, stream>>>(deg_out, deg_in);
    k_wprep<<<(2 * 48 * 64 + 255) / 256, 256, 0, stream>>>(wz, wh, Wzt, Wht);
    int sn = N_EDGES * 16;
    k_scatter<<<(sn + 255) / 256, 256, 0, stream>>>(ei, x, deg_out, deg_in, Txo, Txi);
    k_cell<<<(TILES + 7) / 8, 256, 0, stream>>>(x, Txo, Txi, Wzt, Wht, bz, bh, lw, lb,
                                                (float*)d_out);
}